// Model_89137751262015
// MI455X (gfx1250) — compile-verified
//
#include <hip/hip_runtime.h>
#include <math.h>

typedef __attribute__((ext_vector_type(2))) float v2f;
typedef __attribute__((ext_vector_type(8))) float v8f;

#define SEQ   336
#define PRED  720
#define CH    862
#define CYC   24
#define NB    64
#define CT    16            // channels per workgroup tile
#define WAVES 9
#define TPB   (WAVES * 32)  // 288 threads = 9 wave32
#define MT    5             // M-tiles (16 rows of PRED) per wave; 9*5*16 = 720

// ---------------------------------------------------------------------------
// Prep: K-pair-interleaved operand layouts so each lane's two A values are
// adjacent (one global_load_b64 instead of two b32 in the WMMA hot loop):
//   WTp [s/2][p][2]  : WTp[(s>>1)*2*PRED + 2*p + (s&1)] = W[p][s]
//   S24p[r/2][p][2]  : K=24 cycle-correction operand, same interleaving
//   Wsum[p] = sum_s W[p][s]   (RevIN mean folding)
// ---------------------------------------------------------------------------
__global__ __launch_bounds__(32) void drpk_prep(const float* __restrict__ W,
                                                float* __restrict__ WTp,
                                                float* __restrict__ S24p,
                                                float* __restrict__ Wsum) {
  __shared__ float s24[CYC];
  const int p = blockIdx.x;
  const int t = threadIdx.x;
  if (t < CYC) s24[t] = 0.f;
  __syncthreads();
  for (int s = t; s < SEQ; s += 32) {
    float w = W[p * SEQ + s];
    WTp[(s >> 1) * (2 * PRED) + 2 * p + (s & 1)] = w;
    atomicAdd(&s24[s % CYC], w);
  }
  __syncthreads();
  if (t < CYC) S24p[(t >> 1) * (2 * PRED) + 2 * p + (t & 1)] = s24[t];
  if (t == 0) {
    float acc = 0.f;
    for (int r = 0; r < CYC; ++r) acc += s24[r];
    Wsum[p] = acc;
  }
}

// ---------------------------------------------------------------------------
// Fused main kernel: one workgroup = (batch b, 16-channel tile).
//   pass 0: stage x tile + cycle tile + params in LDS, RevIN stats
//   pass 1: WMMA f32 16x16x4 GEMM (K=336) -> D, K=24 cycle GEMM -> Lc
//   pass 2: per-element epilogue MLP + softmax, LDS-reduce weights over p
//   pass 3: recompute pec from live accumulators, weighted sum, denorm, store
// ---------------------------------------------------------------------------
__global__ __launch_bounds__(TPB) void drpk_main(
    const float* __restrict__ x, const float* __restrict__ x_mark,
    const float* __restrict__ cycle_data, const float* __restrict__ bias,
    const float* __restrict__ W1, const float* __restrict__ b1,
    const float* __restrict__ W2, const float* __restrict__ b2,
    const float* __restrict__ WTp, const float* __restrict__ S24p,
    const float* __restrict__ Wsum, float* __restrict__ out) {
  __shared__ float xs[SEQ * CT];      // 21.5 KB   x tile (time-major)
  __shared__ float cycs[CYC * CT];    //  1.5 KB   cycle tile
  __shared__ float sumS[CT], sqS[CT];
  __shared__ float meanS[CT], stdS[CT], rstdS[CT];
  __shared__ float wsumS[CT * 3];     // softmax-weight reduction over p
  __shared__ float biasS[PRED], wrowS[PRED];
  __shared__ float w1s[3 * 64], b1s[64], w2s[64 * 3], b2s[3];
  __shared__ int kphaseS;

  const int b    = blockIdx.y;
  const int c0   = blockIdx.x * CT;
  const int tid  = threadIdx.x;
  const int wave = tid >> 5;
  const int lane = tid & 31;
  const int cl   = lane & 15;   // column / N within tile
  const int kh   = lane >> 4;   // K-half for A/B operands, M-half for C/D

  if (tid < CT) { sumS[tid] = 0.f; sqS[tid] = 0.f; }
  if (tid < CT * 3) wsumS[tid] = 0.f;
  if (tid == 0) {
    float idx = x_mark[(b * SEQ + (SEQ - 1)) * 4 + 0] * (float)CYC;
    int k = (int)idx;            // idx in [0,24): trunc == floor
    kphaseS = ((k % CYC) + CYC) % CYC;
  }
  for (int i = tid; i < SEQ * CT; i += TPB) {
    int s = i >> 4, c = i & 15, gc = c0 + c;
    xs[i] = (gc < CH) ? x[(b * SEQ + s) * CH + gc] : 0.f;
  }
  for (int i = tid; i < CYC * CT; i += TPB) {
    int m = i >> 4, c = i & 15, gc = c0 + c;
    cycs[i] = (gc < CH) ? cycle_data[m * CH + gc] : 0.f;
  }
  for (int i = tid; i < PRED; i += TPB) { biasS[i] = bias[i]; wrowS[i] = Wsum[i]; }
  for (int i = tid; i < 3 * 64; i += TPB) w1s[i] = W1[i];
  for (int i = tid; i < 64; i += TPB) b1s[i] = b1[i];
  for (int i = tid; i < 64 * 3; i += TPB) w2s[i] = W2[i];
  if (tid < 3) b2s[tid] = b2[tid];
  __syncthreads();

  // RevIN stats: 18 partial reducers per channel, then LDS-atomic combine.
  {
    int c = tid & 15, part = tid >> 4;  // part in [0,18)
    float a = 0.f, a2 = 0.f;
    for (int s = part; s < SEQ; s += TPB / 16) {
      float v = xs[s * CT + c];
      a += v; a2 += v * v;
    }
    atomicAdd(&sumS[c], a);
    atomicAdd(&sqS[c], a2);
  }
  __syncthreads();
  if (tid < CT) {
    float mean = sumS[tid] * (1.f / SEQ);
    float var = (sqS[tid] - (float)SEQ * mean * mean) * (1.f / (SEQ - 1));
    float sd = sqrtf(var + 1e-5f);
    meanS[tid] = mean; stdS[tid] = sd; rstdS[tid] = 1.f / sd;
  }
  __syncthreads();

  const int kph = kphaseS;
  const int m0 = wave * MT * 16;

  v8f accD[MT];   // D = W @ x            (raw, un-normalized)
  v8f accL[MT];   // Lc = W @ cycseq      (cycle-phase correction)
  #pragma unroll
  for (int t = 0; t < MT; ++t) { accD[t] = {}; accL[t] = {}; }

  // Main GEMM: K = 336 in steps of 4 (unrolled x2 -> 10 WMMA + 10 b64 loads
  // + 2 ds_load_2addr per iteration). A from pair-interleaved WTp: one
  // 8-byte load per lane per tile, 128B-coalesced across the 16-lane half.
  // f32 WMMA (16x16x4) preserves exact f32/RNE reference semantics.
  #pragma unroll 2
  for (int k = 0; k < SEQ; k += 4) {
    const int kr = k + 2 * kh;
    v2f bb;
    bb.x = xs[kr * CT + cl];
    bb.y = xs[(kr + 1) * CT + cl];
    const int q = (k >> 1) + kh;          // K-pair row in WTp
    #pragma unroll
    for (int t = 0; t < MT; ++t) {
      const int p = m0 + t * 16 + cl;
      const v2f aa = *(const v2f*)(WTp + (size_t)q * (2 * PRED) + 2 * p);
      accD[t] = __builtin_amdgcn_wmma_f32_16x16x4_f32(
          false, aa, false, bb, (short)0, accD[t], false, false);
    }
  }

  // Cycle-correction GEMM: K = 24; phase rotation folded into B row index.
  #pragma unroll
  for (int kk = 0; kk < CYC; kk += 4) {
    const int jr = kk + 2 * kh;
    v2f bb;
    bb.x = cycs[((jr + kph) % CYC) * CT + cl];
    bb.y = cycs[((jr + 1 + kph) % CYC) * CT + cl];
    const int q = (kk >> 1) + kh;         // K-pair row in S24p
    #pragma unroll
    for (int t = 0; t < MT; ++t) {
      const int p = m0 + t * 16 + cl;
      const v2f aa = *(const v2f*)(S24p + (size_t)q * (2 * PRED) + 2 * p);
      accL[t] = __builtin_amdgcn_wmma_f32_16x16x4_f32(
          false, aa, false, bb, (short)0, accL[t], false, false);
    }
  }

  const float mean = meanS[cl];
  const float rstd = rstdS[cl];
  const float sd   = stdS[cl];
  const int   gc   = c0 + cl;

  // Pass 1: fusion MLP + softmax per (p, c); accumulate weights over p locally
  // (every element this lane owns shares the same channel cl).
  float wl0 = 0.f, wl1 = 0.f, wl2 = 0.f;
  #pragma unroll
  for (int t = 0; t < MT; ++t) {
    #pragma unroll
    for (int j = 0; j < 8; ++j) {
      const int p = m0 + t * 16 + j + 8 * kh;
      const float pred = fmaf(rstd, accD[t][j] - mean * wrowS[p], biasS[p]);
      const float y1 = pred - accL[t][j];
      const float cv = cycs[((p + kph) % CYC) * CT + cl];
      float l0 = b2s[0], l1 = b2s[1], l2 = b2s[2];
      #pragma unroll 1
      for (int h = 0; h < 64; ++h) {
        float hv = fmaf(pred, w1s[h],
                   fmaf(y1, w1s[64 + h],
                   fmaf(cv, w1s[128 + h], b1s[h])));
        hv = fmaxf(hv, 0.f);
        l0 = fmaf(hv, w2s[h * 3 + 0], l0);
        l1 = fmaf(hv, w2s[h * 3 + 1], l1);
        l2 = fmaf(hv, w2s[h * 3 + 2], l2);
      }
      const float mx = fmaxf(l0, fmaxf(l1, l2));
      const float e0 = __expf(l0 - mx);
      const float e1 = __expf(l1 - mx);
      const float e2 = __expf(l2 - mx);
      const float inv = 1.f / (e0 + e1 + e2);
      wl0 = fmaf(e0, inv, wl0);
      wl1 = fmaf(e1, inv, wl1);
      wl2 = fmaf(e2, inv, wl2);
    }
  }
  atomicAdd(&wsumS[cl * 3 + 0], wl0);
  atomicAdd(&wsumS[cl * 3 + 1], wl1);
  atomicAdd(&wsumS[cl * 3 + 2], wl2);
  __syncthreads();

  // Pass 2: gate components with mean-over-p weights, RevIN denorm, store.
  const float wa0 = wsumS[cl * 3 + 0] * (1.f / PRED);
  const float wa1 = wsumS[cl * 3 + 1] * (1.f / PRED);
  const float wa2 = wsumS[cl * 3 + 2] * (1.f / PRED);
  #pragma unroll
  for (int t = 0; t < MT; ++t) {
    #pragma unroll
    for (int j = 0; j < 8; ++j) {
      const int p = m0 + t * 16 + j + 8 * kh;
      const float pred = fmaf(rstd, accD[t][j] - mean * wrowS[p], biasS[p]);
      const float y1 = pred - accL[t][j];
      const float cv = cycs[((p + kph) % CYC) * CT + cl];
      const float y = fmaf(wa0, pred, fmaf(wa1, y1, wa2 * cv));
      if (gc < CH) out[(b * PRED + p) * CH + gc] = fmaf(y, sd, mean);
    }
  }
}

// ---------------------------------------------------------------------------
extern "C" void kernel_launch(void* const* d_in, const int* in_sizes, int n_in,
                              void* d_out, int out_size, void* d_ws,
                              size_t ws_size, hipStream_t stream) {
  (void)in_sizes; (void)n_in; (void)out_size; (void)ws_size;
  const float* x     = (const float*)d_in[0];
  const float* xmark = (const float*)d_in[1];
  const float* cyc   = (const float*)d_in[2];
  const float* W     = (const float*)d_in[3];
  const float* bias  = (const float*)d_in[4];
  const float* W1    = (const float*)d_in[5];
  const float* b1    = (const float*)d_in[6];
  const float* W2    = (const float*)d_in[7];
  const float* b2    = (const float*)d_in[8];
  float* out = (float*)d_out;

  float* WTp  = (float*)d_ws;                 // 336*720 f32 (pair-interleaved)
  float* S24p = WTp + SEQ * PRED;             // 24*720 f32  (pair-interleaved)
  float* Wsum = S24p + CYC * PRED;            // 720 f32 (total ~1.04 MB)

  drpk_prep<<<PRED, 32, 0, stream>>>(W, WTp, S24p, Wsum);
  dim3 grid((CH + CT - 1) / CT, NB);
  drpk_main<<<grid, TPB, 0, stream>>>(x, xmark, cyc, bias, W1, b1, W2, b2,
                                      WTp, S24p, Wsum, out);
}